// ScatterReduceOp_44306882625553
// MI455X (gfx1250) — compile-verified
//
#include <hip/hip_runtime.h>
#include <hip/hip_bf16.h>

// ScatterReduce (sum, include_self): out = input; out[index[i,j], j] += src[i,j]
// Shapes: input [65536,128] f32, index [524288,128] i32, src [524288,128] f32.
// Memory-bound: ~576 MB mandatory traffic -> ~25us @ 23.3 TB/s.
// Strategy: out (32 MB) lives in L2 (192 MB); stream index/src with NT hints;
// resolve the 67M f32 adds with no-return global atomics at L2.

typedef float v4f __attribute__((ext_vector_type(4)));
typedef int   v4i __attribute__((ext_vector_type(4)));

#define D 128

// ---------------------------------------------------------------------------
// Kernel 1: out = input, vectorized 16B copy.
// NT load on input (read exactly once), regular-temporal store on out so the
// destination array is resident in L2 for the atomic phase.
// ---------------------------------------------------------------------------
__global__ __launch_bounds__(256) void
scatter_init_kernel(const float* __restrict__ input,
                    float* __restrict__ out,
                    int n_vec4) {
    const int i = blockIdx.x * blockDim.x + threadIdx.x;
    if (i < n_vec4) {
        const v4f* __restrict__ in4 = (const v4f*)input;
        v4f* __restrict__ out4 = (v4f*)out;
        // input is streamed once: non-temporal load (TH=NT) keeps L2 for `out`
        v4f v = __builtin_nontemporal_load(in4 + i);
        out4[i] = v;   // RT store: keep `out` hot in L2 for the atomics
    }
}

// ---------------------------------------------------------------------------
// Kernel 2: per-element scatter-add.
// One wave32 per source row: lane l owns columns [4l, 4l+3].
//   - int4  load of index (coalesced 512B/wave, NT hint)
//   - float4 load of src   (coalesced 512B/wave, NT hint)
//   - 4 no-return global_atomic_add_f32 into out (random within 32MB, L2-hit)
// global_prefetch_b8 runs ~16KB (32 rows) ahead on both streams.
// ---------------------------------------------------------------------------
__global__ __launch_bounds__(256) void
scatter_add_kernel(const int* __restrict__ index,
                   const float* __restrict__ src,
                   float* __restrict__ out,
                   int n_src_rows) {
    const int tid  = blockIdx.x * blockDim.x + threadIdx.x; // < n_src_rows*32
    const int row  = tid >> 5;          // one wave32 per row
    const int col  = (tid & 31) << 2;   // lane -> 4-column chunk
    if (row >= n_src_rows) return;

    const int base = row * D + col;     // flat element offset (< 2^27, fits i32)

    // Speculative prefetch ~16KB ahead (gfx1250 global_prefetch_b8).
    // OOB at the tail is silently dropped per the ISA (speculative prefetch).
    __builtin_prefetch(src + base + 4096, 0, 1);
    __builtin_prefetch(index + base + 4096, 0, 1);

    // Streamed-once operands: NT loads so they don't evict `out` from L2.
    v4i idx = __builtin_nontemporal_load((const v4i*)(index + base));
    v4f s   = __builtin_nontemporal_load((const v4f*)(src + base));

    // No-return f32 atomics (result unused -> backend emits non-returning
    // global_atomic_add_f32, tracked on STOREcnt only).
    atomicAdd(&out[idx.x * D + (col + 0)], s.x);
    atomicAdd(&out[idx.y * D + (col + 1)], s.y);
    atomicAdd(&out[idx.z * D + (col + 2)], s.z);
    atomicAdd(&out[idx.w * D + (col + 3)], s.w);
}

extern "C" void kernel_launch(void* const* d_in, const int* in_sizes, int n_in,
                              void* d_out, int out_size, void* d_ws, size_t ws_size,
                              hipStream_t stream) {
    const float* input = (const float*)d_in[0];   // [N_DST, 128] f32
    const int*   index = (const int*)  d_in[1];   // [N_SRC, 128] i32
    const float* src   = (const float*)d_in[2];   // [N_SRC, 128] f32
    float*       out   = (float*)d_out;           // [N_DST, 128] f32

    const int n_dst_elems = in_sizes[0];          // N_DST * D
    const int n_src_rows  = in_sizes[2] / D;      // N_SRC

    // Phase 1: out = input (float4-vectorized copy)
    {
        const int n_vec4 = n_dst_elems / 4;       // 2,097,152
        const int block = 256;
        const int grid = (n_vec4 + block - 1) / block;
        scatter_init_kernel<<<grid, block, 0, stream>>>(input, out, n_vec4);
    }

    // Phase 2: atomic scatter-add (one wave32 per source row)
    {
        const int block = 256;                    // 8 wave32s -> 8 rows/block
        const long long total_threads = (long long)n_src_rows * 32;
        const int grid = (int)((total_threads + block - 1) / block); // 65536
        scatter_add_kernel<<<grid, block, 0, stream>>>(index, src, out, n_src_rows);
    }
}